// CLUBLoss_17454747091666
// MI455X (gfx1250) — compile-verified
//
#include <hip/hip_runtime.h>

// ---------------------------------------------------------------------------
// CLUB loss, algebraically collapsed:
//   loss = (-0.5/N) * sum_{i,d} exp(-lv)*( (x-mu)^2 - mu^2 + mu*c[d] - a[d] )
// with a[d] = (sum_j x_jd^2)/N , c[d] = 2*(sum_j x_jd)/N.
// Pure bandwidth problem: ~38.5 MB compulsory traffic -> ~1.7us @ 23.3 TB/s.
// All global traffic is b128; wave reduction uses v_wmma_f32_16x16x4_f32.
// ---------------------------------------------------------------------------

typedef __attribute__((ext_vector_type(2))) float v2f;
typedef __attribute__((ext_vector_type(4))) float v4f;
typedef __attribute__((ext_vector_type(8))) float v8f;

constexpr int Bc = 8, Dc = 512, Hc = 28, Wc = 28;
constexpr int HW  = Hc * Wc;          // 784 (multiple of 4)
constexpr int Nn  = Bc * HW;          // 6272
constexpr int I_TILES = Nn / 4;       // 1568
constexpr int D_TILES = Dc / 4;       // 128
constexpr int NTILES  = I_TILES * D_TILES;  // 200,704 4x4 tiles
constexpr int P2_BLOCKS  = 512;
constexpr int P2_THREADS = 256;

// ---------------------------------------------------------------------------
// Pass 1: column sums of flat_x over the N axis, b128 loads over the
// contiguous h*w axis. Writes ws[d] = s2/N ("a"), ws[D+d] = 2*s1/N ("c").
// ---------------------------------------------------------------------------
__global__ void club_colsum(const float* __restrict__ x, float* __restrict__ ws) {
  const int dd = blockIdx.x;    // 0..511
  const int t  = threadIdx.x;   // 0..255
  constexpr int Q = HW / 4;     // 196 float4s per (bi, dd) plane
  float s1 = 0.f, s2 = 0.f;
  for (int idx = t; idx < Bc * Q; idx += 256) {
    const int bi = idx / Q;
    const int h4 = idx - bi * Q;
    const v4f v = *(const v4f*)(x + (size_t)(bi * Dc + dd) * HW + 4 * h4);
    s1 += (v.x + v.y) + (v.z + v.w);
    s2 = fmaf(v.x, v.x, fmaf(v.y, v.y, fmaf(v.z, v.z, fmaf(v.w, v.w, s2))));
  }
  __shared__ float l1[256], l2[256];
  l1[t] = s1; l2[t] = s2;
  __syncthreads();
  for (int s = 128; s > 0; s >>= 1) {
    if (t < s) { l1[t] += l1[t + s]; l2[t] += l2[t + s]; }
    __syncthreads();
  }
  if (t == 0) {
    ws[dd]      = l2[0] * (1.0f / (float)Nn);   // a[d]
    ws[Dc + dd] = l1[0] * (2.0f / (float)Nn);   // c[d]
  }
}

// ---------------------------------------------------------------------------
// Wave-32 sum via V_WMMA_F32_16X16X4_F32 (documented A 16x4 f32 layout;
// B = all-ones is layout-independent). D columns hold per-half sums; summing
// a lane's 8 D VGPRs + shfl_xor(16) yields the full 32-lane sum.
// ---------------------------------------------------------------------------
__device__ __forceinline__ float wave_sum_wmma(float p) {
  v2f a; a.x = p; a.y = 0.0f;
  v2f b; b.x = 1.0f; b.y = 1.0f;
  v8f c = {};
  v8f d = __builtin_amdgcn_wmma_f32_16x16x4_f32(
      /*neg_a=*/false, a, /*neg_b=*/false, b,
      /*c_mod=*/(short)0, c, /*reuse_a=*/false, /*reuse_b=*/false);
  float s = (((d[0] + d[1]) + (d[2] + d[3])) + ((d[4] + d[5]) + (d[6] + d[7])));
  s += __shfl_xor(s, 16, 32);
  return s;
}

// ---------------------------------------------------------------------------
// Pass 2: fused elementwise + global reduction over 4x4 (i,d) tiles.
//  - p_mu / p_logvar: 4 nontemporal b128 row loads each (contiguous over d),
//    wave-coalesced since consecutive lanes take consecutive d-tiles.
//  - x: 4 b128 loads contiguous over i (the h*w axis); L2-resident (12.8 MB).
//  - a[]/c[]: staged in LDS, read back as b128.
// One deterministic partial per block.
// ---------------------------------------------------------------------------
__global__ void club_main(const float* __restrict__ x,
                          const float* __restrict__ p_mu,
                          const float* __restrict__ p_lv,
                          const float* __restrict__ ac,   // a at [0,D), c at [D,2D)
                          float* __restrict__ partials) {
  __shared__ alignas(16) float sa[Dc];
  __shared__ alignas(16) float sc[Dc];
  __shared__ float wsum[P2_THREADS / 32];

  for (int k = threadIdx.x; k < Dc; k += P2_THREADS) {
    sa[k] = ac[k];
    sc[k] = ac[Dc + k];
  }
  __syncthreads();

  const int tid    = blockIdx.x * P2_THREADS + threadIdx.x;
  const int stride = P2_BLOCKS * P2_THREADS;

  float acc = 0.0f;
  for (int t = tid; t < NTILES; t += stride) {
    const int td  = t & (D_TILES - 1);   // d-tile (fast axis -> mu/lv coalesce)
    const int ti  = t >> 7;              // i-tile
    const int i0  = ti * 4;
    const int dd0 = td * 4;
    const int bi  = i0 / HW;             // i-quad never crosses bi (784 % 4 == 0)
    const int hh  = i0 - bi * HW;

    // x quads: xq[k].r = x element (i0+r, dd0+k)
    const float* xb = x + (size_t)(bi * Dc + dd0) * HW + hh;
    v4f xq[4];
    #pragma unroll
    for (int k = 0; k < 4; ++k) xq[k] = *(const v4f*)(xb + k * HW);

    // mu/lv quads: mu[r][k] = element (i0+r, dd0+k)
    const size_t gb = (size_t)i0 * Dc + dd0;
    v4f mu[4], lv[4];
    #pragma unroll
    for (int r = 0; r < 4; ++r) {
      mu[r] = __builtin_nontemporal_load((const v4f*)(p_mu + gb + (size_t)r * Dc));
      lv[r] = __builtin_nontemporal_load((const v4f*)(p_lv + gb + (size_t)r * Dc));
    }

    const v4f a4 = *(const v4f*)&sa[dd0];
    const v4f c4 = *(const v4f*)&sc[dd0];

    #pragma unroll
    for (int r = 0; r < 4; ++r) {
      #pragma unroll
      for (int k = 0; k < 4; ++k) {
        const float m  = mu[r][k];
        const float xv = xq[k][r];
        const float iv = expf(-lv[r][k]);
        const float dx = xv - m;
        // e = iv * ( dx^2 - a[d] + m*(c[d] - m) )
        const float t0 = fmaf(dx, dx, -a4[k]);
        const float t1 = fmaf(m, c4[k] - m, t0);
        acc = fmaf(iv, t1, acc);
      }
    }
  }

  const float s = wave_sum_wmma(acc);
  const int lane = threadIdx.x & 31;
  const int wid  = threadIdx.x >> 5;
  if (lane == 0) wsum[wid] = s;
  __syncthreads();
  if (threadIdx.x == 0) {
    float tot = 0.0f;
    #pragma unroll
    for (int w = 0; w < P2_THREADS / 32; ++w) tot += wsum[w];
    partials[blockIdx.x] = tot;
  }
}

// ---------------------------------------------------------------------------
// Pass 3: deterministic fixed-order tree over the 512 block partials.
// ---------------------------------------------------------------------------
__global__ void club_finalize(const float* __restrict__ partials,
                              float* __restrict__ out) {
  __shared__ float sm[256];
  const int t = threadIdx.x;
  sm[t] = partials[t] + partials[t + 256];
  __syncthreads();
  for (int s = 128; s > 0; s >>= 1) {
    if (t < s) sm[t] += sm[t + s];
    __syncthreads();
  }
  if (t == 0) out[0] = sm[0] * (-0.5f / (float)Nn);
}

extern "C" void kernel_launch(void* const* d_in, const int* in_sizes, int n_in,
                              void* d_out, int out_size, void* d_ws, size_t ws_size,
                              hipStream_t stream) {
  const float* x    = (const float*)d_in[0];  // (8, 512, 28, 28)
  const float* p_mu = (const float*)d_in[1];  // (6272, 512)
  const float* p_lv = (const float*)d_in[2];  // (6272, 512)
  float* ws  = (float*)d_ws;                  // [0,512): a, [512,1024): c, [1024,1536): partials
  float* out = (float*)d_out;

  club_colsum<<<Dc, 256, 0, stream>>>(x, ws);
  club_main<<<P2_BLOCKS, P2_THREADS, 0, stream>>>(x, p_mu, p_lv, ws, ws + 2 * Dc);
  club_finalize<<<1, 256, 0, stream>>>(ws + 2 * Dc, out);
}